// MolecularGraphEncoder_38886633898028
// MI455X (gfx1250) — compile-verified
//
#include <hip/hip_runtime.h>
#include <hip/hip_bf16.h>

typedef __attribute__((ext_vector_type(2))) float v2f;
typedef __attribute__((ext_vector_type(8))) float v8f;

#define NNODES 50000
#define NEDGES 600000
#define NGRAPHS 1000
#define INDIM 100
#define HIDDIM 128

// -------------------------------------------------------------------------
// C[M,128] = A[M,K] @ W[K,128] (+ optional bias). Row-major everywhere.
// One wave computes a 16-row x 128-col strip using V_WMMA_F32_16X16X4_F32.
// A frag (16x4 f32): lanes 0-15 -> M=row0+lane, K={k,k+1};
//                    lanes 16-31 -> M=row0+lane-16, K={k+2,k+3}.
// B frag (4x16 f32): lanes 0-15 -> N=col0+lane, K={k,k+1};
//                    lanes 16-31 -> N=col0+lane-16, K={k+2,k+3}.
// C/D (16x16 f32): VGPR r -> M=row0+r (lanes 0-15) / M=row0+8+r (lanes 16-31).
// A rows map 1:1 to D rows, so for the (only) partial M-tile we clamp the
// row index instead of zero-filling: garbage rows feed only unstored D rows.
// -------------------------------------------------------------------------
__global__ __launch_bounds__(256) void wmma_gemm128(
    const float* __restrict__ A, const float* __restrict__ W,
    const float* __restrict__ bias, float* __restrict__ C, int M, int K)
{
  const int lane = threadIdx.x & 31;
  const int wave = threadIdx.x >> 5;
  const int mt = blockIdx.x * 8 + wave;           // 16-row strip index
  const int Mtiles = (M + 15) >> 4;
  if (mt >= Mtiles) return;                       // wave-uniform: EXEC stays full
  const int row0 = mt << 4;
  const int half = lane >> 4;
  const int l15  = lane & 15;
  const int m    = min(row0 + l15, M - 1);        // clamped: unconditional loads
  const float* arow = A + (size_t)m * (size_t)K;

  v8f acc[8] = {};

  for (int k0 = 0; k0 < K; k0 += 4) {
    const int kb = k0 + half * 2;
    v2f a = *(const v2f*)(arow + kb);             // one b64 load per lane
    const float* w0 = W + (size_t)kb * 128 + l15;
#pragma unroll
    for (int t = 0; t < 8; ++t) {
      v2f b;
      b.x = w0[t * 16];
      b.y = w0[128 + t * 16];
      acc[t] = __builtin_amdgcn_wmma_f32_16x16x4_f32(
          false, a, false, b, (short)0, acc[t], false, false);
    }
  }

#pragma unroll
  for (int t = 0; t < 8; ++t) {
    const int n = t * 16 + l15;
    const float bv = bias ? bias[n] : 0.0f;
#pragma unroll
    for (int r = 0; r < 8; ++r) {
      const int mm = row0 + r + half * 8;
      if (mm < M) C[(size_t)mm * 128 + n] = acc[t][r] + bv;
    }
  }
}

// ------------------------- degree / normalization -------------------------
__global__ void k_deg_init(float* __restrict__ deg) {
  int i = blockIdx.x * blockDim.x + threadIdx.x;
  if (i < NNODES) deg[i] = 1.0f;                  // +1 self-loop
}
__global__ void k_deg_acc(const int* __restrict__ dst, float* __restrict__ deg) {
  int e = blockIdx.x * blockDim.x + threadIdx.x;
  if (e < NEDGES) atomicAdd(&deg[dst[e]], 1.0f);
}
__global__ void k_rsqrt_inplace(float* __restrict__ deg) {
  int i = blockIdx.x * blockDim.x + threadIdx.x;
  if (i < NNODES) deg[i] = rsqrtf(deg[i]);        // deg -> dinv
}

// ---------------- self-loop contribution + bias (init of agg) -------------
__global__ void k_self_init(const float* __restrict__ h, const float* __restrict__ dinv,
                            const float* __restrict__ bias, float* __restrict__ agg) {
  int tid = blockIdx.x * blockDim.x + threadIdx.x;
  if (tid >= NNODES * 128) return;
  int i = tid >> 7;
  int f = tid & 127;
  float di = dinv[i];
  agg[tid] = h[tid] * di * di + bias[f];
}

// ----------------- edge scatter: one wave per edge, float4/lane -----------
__global__ void k_edge_scatter(const float* __restrict__ h, const int* __restrict__ src,
                               const int* __restrict__ dst, const float* __restrict__ dinv,
                               float* __restrict__ agg) {
  int tid = blockIdx.x * blockDim.x + threadIdx.x;
  int e = tid >> 5;
  int lane = tid & 31;
  if (e >= NEDGES) return;
  int s = src[e], d = dst[e];
  float norm = dinv[s] * dinv[d];
  float4 v = ((const float4*)(h + (size_t)s * 128))[lane];
  float* ad = agg + (size_t)d * 128 + lane * 4;
  atomicAdd(ad + 0, v.x * norm);
  atomicAdd(ad + 1, v.y * norm);
  atomicAdd(ad + 2, v.z * norm);
  atomicAdd(ad + 3, v.w * norm);
}

__global__ void k_relu4(float* __restrict__ x, int n4) {
  int i = blockIdx.x * blockDim.x + threadIdx.x;
  if (i >= n4) return;
  float4* p = (float4*)x;
  float4 v = p[i];
  v.x = fmaxf(v.x, 0.0f);
  v.y = fmaxf(v.y, 0.0f);
  v.z = fmaxf(v.z, 0.0f);
  v.w = fmaxf(v.w, 0.0f);
  p[i] = v;
}

// ------------------------------- pooling ----------------------------------
__global__ void k_zero(float* __restrict__ p, int n) {
  int i = blockIdx.x * blockDim.x + threadIdx.x;
  if (i < n) p[i] = 0.0f;
}
__global__ void k_pool_acc(const float* __restrict__ h, const int* __restrict__ batch,
                           float* __restrict__ pooled, float* __restrict__ cnt) {
  int tid = blockIdx.x * blockDim.x + threadIdx.x;
  int node = tid >> 5;
  int lane = tid & 31;
  if (node >= NNODES) return;
  int g = batch[node];
  if (lane == 0) atomicAdd(&cnt[g], 1.0f);
  float4 v = ((const float4*)(h + (size_t)node * 128))[lane];
  float* pg = pooled + (size_t)g * 128 + lane * 4;
  atomicAdd(pg + 0, v.x);
  atomicAdd(pg + 1, v.y);
  atomicAdd(pg + 2, v.z);
  atomicAdd(pg + 3, v.w);
}
__global__ void k_pool_div(float* __restrict__ pooled, const float* __restrict__ cnt) {
  int tid = blockIdx.x * blockDim.x + threadIdx.x;
  if (tid >= NGRAPHS * 128) return;
  int g = tid >> 7;
  pooled[tid] *= 1.0f / fmaxf(cnt[g], 1.0f);
}

// --------------------------------------------------------------------------
extern "C" void kernel_launch(void* const* d_in, const int* in_sizes, int n_in,
                              void* d_out, int out_size, void* d_ws, size_t ws_size,
                              hipStream_t stream) {
  const float* x    = (const float*)d_in[0];
  const int* ei     = (const int*)d_in[1];   // [2, NEDGES] row-major
  const int* src    = ei;
  const int* dst    = ei + NEDGES;
  const int* batch  = (const int*)d_in[2];
  const float* W1   = (const float*)d_in[3];
  const float* b1   = (const float*)d_in[4];
  const float* W2   = (const float*)d_in[5];
  const float* b2   = (const float*)d_in[6];
  const float* Wp   = (const float*)d_in[7];
  const float* bp   = (const float*)d_in[8];
  float* out = (float*)d_out;

  float* ws     = (float*)d_ws;
  float* hA     = ws;                                   // [NNODES,128]
  float* hB     = hA + (size_t)NNODES * 128;            // [NNODES,128]
  float* dinv   = hB + (size_t)NNODES * 128;            // [NNODES]
  float* pooled = dinv + NNODES;                        // [NGRAPHS,128]
  float* cnt    = pooled + (size_t)NGRAPHS * 128;       // [NGRAPHS] (contig after pooled)

  const int B = 256;
  auto cdiv = [](long a, long b) { return (int)((a + b - 1) / b); };

  // 1) symmetric normalization factors
  k_deg_init<<<cdiv(NNODES, B), B, 0, stream>>>(dinv);
  k_deg_acc<<<cdiv(NEDGES, B), B, 0, stream>>>(dst, dinv);
  k_rsqrt_inplace<<<cdiv(NNODES, B), B, 0, stream>>>(dinv);

  const int gemmBlocksN = cdiv((NNODES + 15) / 16, 8);

  // 2) layer 1: hA = x @ W1 ; hB = aggregate + self-loop + b1 ; relu(hB)
  wmma_gemm128<<<gemmBlocksN, B, 0, stream>>>(x, W1, nullptr, hA, NNODES, INDIM);
  k_self_init<<<cdiv((long)NNODES * 128, B), B, 0, stream>>>(hA, dinv, b1, hB);
  k_edge_scatter<<<cdiv((long)NEDGES * 32, B), B, 0, stream>>>(hA, src, dst, dinv, hB);
  k_relu4<<<cdiv((long)NNODES * 32, B), B, 0, stream>>>(hB, NNODES * 32);

  // 3) layer 2: hA = hB @ W2 ; hB = aggregate + self-loop + b2 ; relu(hB)
  wmma_gemm128<<<gemmBlocksN, B, 0, stream>>>(hB, W2, nullptr, hA, NNODES, HIDDIM);
  k_self_init<<<cdiv((long)NNODES * 128, B), B, 0, stream>>>(hA, dinv, b2, hB);
  k_edge_scatter<<<cdiv((long)NEDGES * 32, B), B, 0, stream>>>(hA, src, dst, dinv, hB);
  k_relu4<<<cdiv((long)NNODES * 32, B), B, 0, stream>>>(hB, NNODES * 32);

  // 4) global mean pool
  k_zero<<<cdiv(NGRAPHS * 128 + NGRAPHS, B), B, 0, stream>>>(pooled, NGRAPHS * 128 + NGRAPHS);
  k_pool_acc<<<cdiv((long)NNODES * 32, B), B, 0, stream>>>(hB, batch, pooled, cnt);
  k_pool_div<<<cdiv(NGRAPHS * 128, B), B, 0, stream>>>(pooled, cnt);

  // 5) head: out = pooled @ Wp + bp
  wmma_gemm128<<<cdiv((NGRAPHS + 15) / 16, 8), B, 0, stream>>>(pooled, Wp, bp, out, NGRAPHS, HIDDIM);
}